// surrogate_gradient_adder_dense_53111565582738
// MI455X (gfx1250) — compile-verified
//
#include <hip/hip_runtime.h>

typedef __attribute__((ext_vector_type(2))) float v2f;
typedef __attribute__((ext_vector_type(4))) float v4f;
typedef __attribute__((ext_vector_type(8))) float v8f;
typedef __attribute__((ext_vector_type(4))) int   v4i;

typedef __attribute__((address_space(1))) v4i* gptr_v4i;  // global
typedef __attribute__((address_space(3))) v4i* lptr_v4i;  // LDS

#define ROWF 20  // padded LDS row stride (16 data + 4 pad floats): conflict-free column reads, 16B-aligned rows

__device__ __forceinline__ unsigned long long spread16to64(unsigned int x) {
  // deposit 16 bits into every 4th bit of a 64-bit word (nibble-per-bit counters)
  unsigned long long v = x;
  v = (v | (v << 24)) & 0x000000ff000000ffull;
  v = (v | (v << 12)) & 0x000f000f000f000full;
  v = (v | (v << 6))  & 0x0303030303030303ull;
  v = (v | (v << 3))  & 0x1111111111111111ull;
  return v;
}

__global__ __launch_bounds__(256) void sgad_kernel(const float* __restrict__ x,
                                                   float* __restrict__ out,
                                                   int B) {
#if defined(__HIP_DEVICE_COMPILE__)
  __shared__ float tile[8][16 * ROWF];
  const int tid   = threadIdx.x;
  const int wave  = tid >> 5;
  const int lane  = tid & 31;
  const int batch = blockIdx.x * 8 + wave;
  if (batch >= B) return;

  float* wlds = tile[wave];
  const int h = lane >> 4;  // wave half (0/1)
  const int m = lane & 15;  // row index within half (WMMA A layout)

  // ---------- stage 1 KB batch tile: global -> LDS (padded rows) ----------
  {
    const int l = lane >> 1;  // destination row 0..15
    const int j = lane & 1;   // which 8-float half of the row
    const float* g = x + (size_t)batch * 256 + (size_t)lane * 8;
    float* s = wlds + l * ROWF + j * 8;
#if __has_builtin(__builtin_amdgcn_global_load_async_to_lds_b128)
    __builtin_amdgcn_global_load_async_to_lds_b128(
        (gptr_v4i)g, (lptr_v4i)s, 0, 0);
    __builtin_amdgcn_global_load_async_to_lds_b128(
        (gptr_v4i)(g + 4), (lptr_v4i)(s + 4), 0, 0);
#if __has_builtin(__builtin_amdgcn_s_wait_asynccnt)
    __builtin_amdgcn_s_wait_asynccnt(0);
#else
    asm volatile("s_wait_asynccnt 0" ::: "memory");
#endif
#else
    v4f r0 = *(const v4f*)g;
    v4f r1 = *(const v4f*)(g + 4);
    *(v4f*)s = r0;
    *(v4f*)(s + 4) = r1;
    asm volatile("" ::: "memory");
#endif
  }

  // ---------- x_int via WMMA f32 16x16x4 (4 chunks over K=16) ----------
  // A layout: lane (m,h): VGPR0 = X[m, 4c+2h], VGPR1 = X[m, 4c+2h+1]
  // B = W with W[k,n] = 2^k (broadcast over n)  =>  D[l,n] = x_int[l] for all n
  const float* arow = wlds + m * ROWF + 2 * h;
  v2f a0 = *(const v2f*)(arow + 0);
  v2f a1 = *(const v2f*)(arow + 4);
  v2f a2 = *(const v2f*)(arow + 8);
  v2f a3 = *(const v2f*)(arow + 12);

  v2f b0 = {(float)(1u << (0 + 2 * h)),  (float)(1u << (1 + 2 * h))};
  v2f b1 = {(float)(1u << (4 + 2 * h)),  (float)(1u << (5 + 2 * h))};
  v2f b2 = {(float)(1u << (8 + 2 * h)),  (float)(1u << (9 + 2 * h))};
  v2f b3 = {(float)(1u << (12 + 2 * h)), (float)(1u << (13 + 2 * h))};

  v8f acc = {0.f, 0.f, 0.f, 0.f, 0.f, 0.f, 0.f, 0.f};
  acc = __builtin_amdgcn_wmma_f32_16x16x4_f32(false, a0, false, b0, (short)0, acc, false, false);
  acc = __builtin_amdgcn_wmma_f32_16x16x4_f32(false, a1, false, b1, (short)0, acc, false, false);
  acc = __builtin_amdgcn_wmma_f32_16x16x4_f32(false, a2, false, b2, (short)0, acc, false, false);
  acc = __builtin_amdgcn_wmma_f32_16x16x4_f32(false, a3, false, b3, (short)0, acc, false, false);
  // now acc[r] == x_int[r + 8*h] (identical across the 16 lanes of each half)

  // ---------- x_sum ----------
  float shalf = acc[0] + acc[1] + acc[2] + acc[3] + acc[4] + acc[5] + acc[6] + acc[7];
  float xsum = shalf + __shfl_xor(shalf, 16, 32);

  // ---------- packed rounded bits of each row (abits) ----------
  unsigned p = 0;
  p |= ((unsigned)(int)rintf(a0.x)) << (0 + 2 * h);
  p |= ((unsigned)(int)rintf(a0.y)) << (1 + 2 * h);
  p |= ((unsigned)(int)rintf(a1.x)) << (4 + 2 * h);
  p |= ((unsigned)(int)rintf(a1.y)) << (5 + 2 * h);
  p |= ((unsigned)(int)rintf(a2.x)) << (8 + 2 * h);
  p |= ((unsigned)(int)rintf(a2.y)) << (9 + 2 * h);
  p |= ((unsigned)(int)rintf(a3.x)) << (12 + 2 * h);
  p |= ((unsigned)(int)rintf(a3.y)) << (13 + 2 * h);
  unsigned afull = p | (unsigned)__shfl_xor((int)p, 16, 32);  // lane m holds abits[m]

  // ---------- collapsed ripple-carry add + per-bit counting ----------
  unsigned long long accS = 0ull, accC = 0ull;
#pragma unroll
  for (int r = 0; r < 8; ++r) {
    unsigned ar = (unsigned)__shfl((int)afull, r + 8 * h, 32);  // abits[r+8h]
    int resid = (int)(xsum - acc[r]);                            // trunc like .astype(int32)
    unsigned br = ((unsigned)resid) & 0xFFFFu;
    unsigned full = ar + br;
    unsigned sbits = full & 0xFFFFu;                             // ripple sum bits
    unsigned cbits = ((full ^ ar ^ br) >> 1) & 0xFFFFu;          // per-bit carry-outs
    accS += spread16to64(sbits);
    accC += spread16to64(cbits);
  }

  // ---------- extract per-bit counts, combine halves, store ----------
  unsigned cs = (unsigned)((accS >> (4 * m)) & 0xFull);
  unsigned cc = (unsigned)((accC >> (4 * m)) & 0xFull);
  int ts = (int)cs + __shfl_xor((int)cs, 16, 32);  // sum-bit count over all 16 rows
  int tc = (int)cc + __shfl_xor((int)cc, 16, 32);  // carry-bit count over all 16 rows

  // lanes 0..15 write out (mean = count/16), lanes 16..31 write carry (sum)
  float val = (h == 0) ? ((float)ts * 0.0625f) : (float)tc;
  size_t off = (h == 0) ? ((size_t)batch * 16 + (size_t)m)
                        : ((size_t)B * 16 + (size_t)batch * 16 + (size_t)m);
  out[off] = val;
#endif  // __HIP_DEVICE_COMPILE__
}

extern "C" void kernel_launch(void* const* d_in, const int* in_sizes, int n_in,
                              void* d_out, int out_size, void* d_ws, size_t ws_size,
                              hipStream_t stream) {
  (void)n_in; (void)out_size; (void)d_ws; (void)ws_size;
  const float* x = (const float*)d_in[0];
  float* out = (float*)d_out;
  const int B = in_sizes[0] / 256;        // (B,16,16) float tiles
  const int blocks = (B + 7) / 8;         // one batch per wave, 8 waves (256 thr) per block
  sgad_kernel<<<blocks, 256, 0, stream>>>(x, out, B);
}